// E74LowRankCell_23751169147416
// MI455X (gfx1250) — compile-verified
//
#include <hip/hip_runtime.h>
#include <math.h>

// ---------------------------------------------------------------------------
// E74 low-rank cell for MI455X (gfx1250, wave32)
//   Phase 1: k/v/q projections via native fp32 WMMA (v_wmma_f32_16x16x4_f32)
//            wave tile 32(M) x 64(N): 8 WMMAs per 6 b64 loads (~10.7 FLOP/B)
//   Phase 2: sequential scan, one workgroup per batch, fused 25-way reduction
// Workspace: 3 * T*B*n * 4B = 192 MB for k/v/q.
// ---------------------------------------------------------------------------

typedef __attribute__((ext_vector_type(2))) float v2f;
typedef __attribute__((ext_vector_type(8))) float v8f;

#define T_DIM 512
#define B_DIM 32
#define D_DIM 1024
#define N_DIM 1024
#define R_DIM 8
#define EPS_F 1e-6f

// ===========================================================================
// Phase 1: C[m, n] = sum_d x[m, d] * W[n, d]   (m = t*B + b)
// Block = 256 threads (8 waves). Each wave computes a 32x64 output tile:
// 2 A fragments (M=+0, M=+16) x 4 B fragments (N strips of 16) = 8 accums.
// A fragment (16x4 f32): lane L<16 holds A[M=L, K=k0..k0+1], lane L>=16 holds
// A[M=L-16, K=k0+2..k0+3]  -> one float2 load per lane. B (=W^T) identical.
// ===========================================================================
__global__ __launch_bounds__(256)
void e74_proj_gemm(const float* __restrict__ x,
                   const float* __restrict__ Wk,
                   const float* __restrict__ Wv,
                   const float* __restrict__ Wq,
                   float* __restrict__ kbuf,
                   float* __restrict__ vbuf,
                   float* __restrict__ qbuf)
{
    const int lane   = threadIdx.x & 31;
    const int wave   = threadIdx.x >> 5;
    const int m_base = blockIdx.x * 256 + wave * 32;   // 16384 / 256 = 64 blocks
    const int n_base = blockIdx.y * 64;                // 1024 / 64 = 16 blocks

    const float* __restrict__ W;
    float* __restrict__ dst;
    if (blockIdx.z == 0)      { W = Wk; dst = kbuf; }
    else if (blockIdx.z == 1) { W = Wv; dst = vbuf; }
    else                      { W = Wq; dst = qbuf; }

    const int lo16  = lane & 15;
    const int khalf = 2 * (lane >> 4);                 // 0 or 2

    const float* a_ptr = x + (size_t)(m_base + lo16) * D_DIM + khalf;
    const float* b_ptr = W + (size_t)(n_base + lo16) * D_DIM + khalf;

    v8f acc[8];
    const v8f zero8 = {0.f, 0.f, 0.f, 0.f, 0.f, 0.f, 0.f, 0.f};
#pragma unroll
    for (int i = 0; i < 8; ++i) acc[i] = zero8;

#pragma unroll 2
    for (int k0 = 0; k0 < D_DIM; k0 += 4) {
        v2f a0 = *(const v2f*)(a_ptr + k0);
        v2f a1 = *(const v2f*)(a_ptr + 16 * D_DIM + k0);
        v2f b0 = *(const v2f*)(b_ptr + k0);
        v2f b1 = *(const v2f*)(b_ptr + 16 * D_DIM + k0);
        v2f b2 = *(const v2f*)(b_ptr + 32 * D_DIM + k0);
        v2f b3 = *(const v2f*)(b_ptr + 48 * D_DIM + k0);
        acc[0] = __builtin_amdgcn_wmma_f32_16x16x4_f32(
                    false, a0, false, b0, (short)0, acc[0], false, false);
        acc[1] = __builtin_amdgcn_wmma_f32_16x16x4_f32(
                    false, a0, false, b1, (short)0, acc[1], false, false);
        acc[2] = __builtin_amdgcn_wmma_f32_16x16x4_f32(
                    false, a0, false, b2, (short)0, acc[2], false, false);
        acc[3] = __builtin_amdgcn_wmma_f32_16x16x4_f32(
                    false, a0, false, b3, (short)0, acc[3], false, false);
        acc[4] = __builtin_amdgcn_wmma_f32_16x16x4_f32(
                    false, a1, false, b0, (short)0, acc[4], false, false);
        acc[5] = __builtin_amdgcn_wmma_f32_16x16x4_f32(
                    false, a1, false, b1, (short)0, acc[5], false, false);
        acc[6] = __builtin_amdgcn_wmma_f32_16x16x4_f32(
                    false, a1, false, b2, (short)0, acc[6], false, false);
        acc[7] = __builtin_amdgcn_wmma_f32_16x16x4_f32(
                    false, a1, false, b3, (short)0, acc[7], false, false);
    }

    // C layout: VGPR j, lane L -> row = 8*(L/16)+j, col = L%16
    const int c_col0 = n_base + lo16;
#pragma unroll
    for (int half = 0; half < 2; ++half) {             // M=+0 block, M=+16 block
        const int c_row0 = m_base + half * 16 + 8 * (lane >> 4);
#pragma unroll
        for (int i = 0; i < 4; ++i) {
            const v8f a = acc[half * 4 + i];
#pragma unroll
            for (int j = 0; j < 8; ++j) {
                dst[(size_t)(c_row0 + j) * N_DIM + (c_col0 + i * 16)] = a[j];
            }
        }
    }
}

// ===========================================================================
// Phase 2: sequential scan. One workgroup per batch b; 1024 threads, thread
// owns row `n = threadIdx.x` of U, V, and column n of W_kr (all in VGPRs).
// Per step one fused 25-value reduction: [ |k|^2, V^T k (8), W_kr k (8),
// V^T q (8) ] — all linear in un-normalized k, scale applied after sqrt.
// ===========================================================================
__global__ __launch_bounds__(1024)
void e74_scan(const float* __restrict__ kbuf,
              const float* __restrict__ vbuf,
              const float* __restrict__ qbuf,
              const float* __restrict__ W_kr,
              const float* __restrict__ U0,
              const float* __restrict__ V0,
              float* __restrict__ out,     // [T, B, N]
              float* __restrict__ U_out,   // [B, N, R]
              float* __restrict__ V_out)   // [B, N, R]
{
    const int b    = blockIdx.x;
    const int row  = threadIdx.x;       // 0..1023
    const int lane = threadIdx.x & 31;
    const int wave = threadIdx.x >> 5;  // 0..31

    float U[R_DIM], V[R_DIM], Wr[R_DIM];
    const size_t uvoff = ((size_t)b * N_DIM + row) * R_DIM;
#pragma unroll
    for (int j = 0; j < R_DIM; ++j) {
        U[j]  = U0[uvoff + j];
        V[j]  = V0[uvoff + j];
        Wr[j] = W_kr[(size_t)j * N_DIM + row];
    }

    __shared__ float red[32][28];   // per-wave partials (25 used, padded)
    __shared__ float fin[25];       // block-wide sums

    const size_t step_stride = (size_t)B_DIM * N_DIM;

    for (int t = 0; t < T_DIM; ++t) {
        const size_t off = ((size_t)t * B_DIM + b) * N_DIM + row;
        const float kt = kbuf[off];
        const float vt = vbuf[off];
        const float qt = qbuf[off];

        if (t + 1 < T_DIM) {        // pull next step toward the WGP early
            __builtin_prefetch(kbuf + off + step_stride, 0, 0);
            __builtin_prefetch(vbuf + off + step_stride, 0, 0);
            __builtin_prefetch(qbuf + off + step_stride, 0, 0);
        }

        // ---- partials ----
        float p[25];
        p[0] = kt * kt;
#pragma unroll
        for (int j = 0; j < R_DIM; ++j) {
            p[1 + j]  = V[j]  * kt;   // V^T k   (unnormalized)
            p[9 + j]  = Wr[j] * kt;   // W_kr k  (unnormalized)
            p[17 + j] = V[j]  * qt;   // V^T q
        }

        // ---- wave32 xor-tree reduction ----
#pragma unroll
        for (int m = 16; m >= 1; m >>= 1) {
#pragma unroll
            for (int i = 0; i < 25; ++i)
                p[i] += __shfl_xor(p[i], m, 32);
        }
        if (lane == 0) {
#pragma unroll
            for (int i = 0; i < 25; ++i) red[wave][i] = p[i];
        }
        __syncthreads();

        // ---- cross-wave sum (threads 0..24, one value each) ----
        if (threadIdx.x < 25) {
            float s = 0.f;
#pragma unroll
            for (int w = 0; w < 32; ++w) s += red[w][threadIdx.x];
            fin[threadIdx.x] = s;
        }
        __syncthreads();

        const float scale = 1.0f / (sqrtf(fin[0]) + EPS_F);
        float Vtk[R_DIM], kr[R_DIM], Vtq[R_DIM];
#pragma unroll
        for (int j = 0; j < R_DIM; ++j) {
            Vtk[j] = fin[1 + j]  * scale;
            kr[j]  = fin[9 + j]  * scale;
            Vtq[j] = fin[17 + j];
        }

        // ---- per-row state update ----
        float retrieved = 0.f;
#pragma unroll
        for (int j = 0; j < R_DIM; ++j) retrieved += U[j] * Vtk[j];
        const float delta = vt - retrieved;
#pragma unroll
        for (int j = 0; j < R_DIM; ++j) U[j] = tanhf(U[j] + delta * kr[j]);

        float Sq = 0.f;
#pragma unroll
        for (int j = 0; j < R_DIM; ++j) Sq += U[j] * Vtq[j];

        const float sig = 1.0f / (1.0f + __expf(-Sq));
        out[off] = Sq * Sq * sig;     // Sq * silu(Sq)
    }

#pragma unroll
    for (int j = 0; j < R_DIM; ++j) {
        U_out[uvoff + j] = U[j];
        V_out[uvoff + j] = V[j];      // V never updated: copy of V0
    }
}

// ===========================================================================
extern "C" void kernel_launch(void* const* d_in, const int* in_sizes, int n_in,
                              void* d_out, int out_size, void* d_ws, size_t ws_size,
                              hipStream_t stream)
{
    (void)in_sizes; (void)n_in; (void)out_size; (void)ws_size;

    const float* x   = (const float*)d_in[0];
    const float* Wk  = (const float*)d_in[1];
    const float* Wv  = (const float*)d_in[2];
    const float* Wq  = (const float*)d_in[3];
    const float* Wkr = (const float*)d_in[4];
    const float* U0  = (const float*)d_in[5];
    const float* V0  = (const float*)d_in[6];

    const size_t TBN = (size_t)T_DIM * B_DIM * N_DIM;   // 16,777,216
    float* kbuf = (float*)d_ws;
    float* vbuf = kbuf + TBN;
    float* qbuf = vbuf + TBN;

    float* out   = (float*)d_out;
    float* U_out = out + TBN;
    float* V_out = U_out + (size_t)B_DIM * N_DIM * R_DIM;

    // Phase 1: projections (k, v, q) via fp32 WMMA
    dim3 grid_g((T_DIM * B_DIM) / 256, N_DIM / 64, 3);  // (64, 16, 3)
    e74_proj_gemm<<<grid_g, 256, 0, stream>>>(x, Wk, Wv, Wq, kbuf, vbuf, qbuf);

    // Phase 2: sequential scan, one workgroup per batch
    e74_scan<<<B_DIM, 1024, 0, stream>>>(kbuf, vbuf, qbuf, Wkr, U0, V0,
                                         out, U_out, V_out);
}